// SparseKnnGraph_11433202942846
// MI455X (gfx1250) — compile-verified
//
#include <hip/hip_runtime.h>

#define B_  2
#define N_  8192
#define C_  256
#define K_  9
#define KK  10          // K+1 kept per row
#define NCHUNK 8        // column chunks per row
#define CHUNK 1024      // columns per chunk (per block)
#define MT 64           // rows per block tile
#define NT 128          // cols per sub-tile
#define KS 32           // K slice depth staged in LDS (double buffered)
#define NKS (C_ / KS)   // 8 K slices
#define LDP 36          // padded LDS pitch for A/B slices: 32 DWORDs + 4 pad
#define LDD 132         // padded LDS pitch for dist tile

// LDS layout, float offsets (byte offsets = x4):
//   As0 @ 0      (2304)   As1 @ 2304  (2304)
//   Bs0 @ 4608   (4608)   Bs1 @ 9216  (4608)
//   Ds  @ 4608 (alias Bs region, 64x132 = 8448 <= 9216)
//   Md  @ 0 (2560) / Mi @ 2560 (alias As region, merge scratch)
//   sqs @ 13824 (persistent, 64)
#define AOFF0_F 0
#define AOFF1_F 2304
#define BOFF0_F 4608
#define BOFF1_F 9216
#define SQOFF_F 13824
#define SMEM_F  13888

typedef float v2f __attribute__((ext_vector_type(2)));
typedef float v8f __attribute__((ext_vector_type(8)));
typedef unsigned v4u __attribute__((ext_vector_type(4)));
typedef int v8i __attribute__((ext_vector_type(8)));
typedef int v4i __attribute__((ext_vector_type(4)));

// ---- Tensor Data Mover: async 2D tile DMA global->LDS with row padding -----
// D# packing per CDNA5 ISA ch.8: group0 = {count/lds_addr/global_addr/type},
// group1 = {flags, tensor dims, tile dims, dim0 stride}. 2D tile: groups 2/3
// zero. data_size code 2 = 4B elems. pad: every 32 DWORDs insert 4 DWORDs ->
// LDS tile lands at pitch LDP=36 floats (conflict-spread WMMA frag reads).
// 6-arg builtin on this toolchain:
//   (uint32x4 g0, int32x8 g1, int32x4 g2, int32x4 g3, int32x8, i32 cpol)
__device__ __forceinline__ void tdm_load_2d(unsigned lds_off_bytes,
                                            const float* gaddr,
                                            unsigned tile_d1 /*rows*/) {
  unsigned long long ga = (unsigned long long)(__SIZE_TYPE__)gaddr;
  v4u g0;
  g0[0] = 1u;                                          // count=1, user D#
  g0[1] = lds_off_bytes;                               // lds_addr
  g0[2] = (unsigned)(ga & 0xFFFFFFFFu);                // global_addr[31:0]
  g0[3] = (unsigned)((ga >> 32) & 0x1FFFFFFu)          // global_addr[56:32]
        | (2u << 30);                                  // type=2 (image)
  v8i g1;
  g1[0] = (int)((2u << 16)        // data_size = 4 bytes
              | (1u << 20)        // pad_enable
              | (4u << 22)        // pad_interval code 4 = 32 DWORDs
              | (3u << 25));      // pad_amount  code 3 = 4 DWORDs
  g1[1] = (int)(256u << 16);                       // tensor_dim0 = 256 (lo16)
  g1[2] = (int)((256u >> 16) | (8192u << 16));     // tensor_dim0 hi | dim1 lo
  g1[3] = (int)((8192u >> 16) | ((unsigned)KS << 16)); // dim1 hi | tile_dim0=32
  g1[4] = (int)(tile_d1 & 0xFFFFu);                // tile_dim1 rows, tile_dim2=0
  g1[5] = (int)256u;                               // tensor_dim0_stride = C
  g1[6] = 0;
  g1[7] = 0;
  v4i z4 = {0, 0, 0, 0};
  v8i z8 = {0, 0, 0, 0, 0, 0, 0, 0};
  __builtin_amdgcn_tensor_load_to_lds(g0, g1, z4, z4, z8, 0);
}

// ---------------- Kernel 1: row L2-normalize + sq = sum(xn^2) ----------------
__global__ void __launch_bounds__(256) knn_normalize(const float* __restrict__ x,
                                                     float* __restrict__ xn,
                                                     float* __restrict__ sq) {
  __shared__ float red[256];
  const int g = blockIdx.x;            // row 0 .. B*N-1
  const int t = threadIdx.x;           // 0 .. 255 == C
  float v = x[(long)g * C_ + t];
  red[t] = v * v;
  __syncthreads();
  for (int s = 128; s > 0; s >>= 1) {
    if (t < s) red[t] += red[t + s];
    __syncthreads();
  }
  float norm = sqrtf(red[0]);
  __syncthreads();
  float denom = fmaxf(norm, 1e-12f);
  float xv = v / denom;
  xn[(long)g * C_ + t] = xv;
  red[t] = xv * xv;
  __syncthreads();
  for (int s = 128; s > 0; s >>= 1) {
    if (t < s) red[t] += red[t + s];
    __syncthreads();
  }
  if (t == 0) sq[g] = red[0];
}

// -- Kernel 2: TDM double-buffered WMMA-f32 GEMM tile + streaming top-10 -----
__global__ void __launch_bounds__(256) knn_gemm_topk(const float* __restrict__ xn,
                                                     const float* __restrict__ sq,
                                                     float* __restrict__ cand_d,
                                                     int*   __restrict__ cand_i) {
  __shared__ float smem[SMEM_F];   // 55,552 bytes static LDS
  float* Ds  = smem + BOFF0_F;
  float* Md  = smem;
  int*   Mi  = (int*)(smem + 2560);
  float* sqs = smem + SQOFF_F;

  const int t    = threadIdx.x;
  const int w    = t >> 5;           // wave 0..7
  const int lane = t & 31;
  const int hi   = lane >> 4;        // K-half select for f32 WMMA frags
  const int ln   = lane & 15;

  const int rowtile = blockIdx.x;    // 0..127
  const int chunk   = blockIdx.y;    // 0..7
  const int b       = blockIdx.z;    // 0..1
  const int rowbase = rowtile * MT;
  const int colbase = chunk * CHUNK;
  const long gb     = (long)b * N_;

  // Runtime LDS byte offset of smem (low 32 bits of the generic LDS address
  // are the in-group offset). Kept as a runtime value so nothing here gets
  // folded into a static initializer.
  const unsigned smemBase = (unsigned)(__SIZE_TYPE__)(void*)smem;

  if (t < MT) sqs[t] = sq[gb + rowbase + t];

  float bd[KK];
  int   bi[KK];
#pragma unroll
  for (int s = 0; s < KK; ++s) { bd[s] = 3.4e38f; bi[s] = 0x7fffffff; }

  for (int sub = 0; sub < CHUNK / NT; ++sub) {
    const int subcol = colbase + sub * NT;

    v8f acc[4];
#pragma unroll
    for (int mt = 0; mt < 4; ++mt)
#pragma unroll
      for (int r = 0; r < 8; ++r) acc[mt][r] = 0.0f;

    // Preload K-slice 0 into buffer 0 (wave 0 drives the TDM).
    __syncthreads();  // previous sub's readers of the aliased LDS are done
    if (w == 0) {
      tdm_load_2d(smemBase + AOFF0_F * 4u, &xn[(gb + rowbase) * C_ + 0], MT);
      tdm_load_2d(smemBase + BOFF0_F * 4u, &xn[(gb + subcol) * C_ + 0], NT);
      __builtin_amdgcn_s_wait_tensorcnt(0);
    }
    __syncthreads();

    for (int ks = 0; ks < NKS; ++ks) {
      const int cur = ks & 1;
      // Kick off next slice's DMA before computing this one.
      if (w == 0 && ks + 1 < NKS) {
        const int kb = (ks + 1) * KS;
        unsigned aDst = smemBase + (cur ? AOFF0_F : AOFF1_F) * 4u;
        unsigned bDst = smemBase + (cur ? BOFF0_F : BOFF1_F) * 4u;
        tdm_load_2d(aDst, &xn[(gb + rowbase) * C_ + kb], MT);
        tdm_load_2d(bDst, &xn[(gb + subcol) * C_ + kb], NT);
      }
      const float* Ac = smem + (cur ? AOFF1_F : AOFF0_F);
      const float* Bc = smem + (cur ? BOFF1_F : BOFF0_F);
      // 8 k-steps of V_WMMA_F32_16X16X4_F32 on the current buffers.
#pragma unroll 4
      for (int kk = 0; kk < KS / 4; ++kk) {
        int k0 = kk * 4 + 2 * hi;
        v2f bfrag = *(const v2f*)&Bc[(w * 16 + ln) * LDP + k0];
#pragma unroll
        for (int mt = 0; mt < 4; ++mt) {
          v2f afrag = *(const v2f*)&Ac[(mt * 16 + ln) * LDP + k0];
          acc[mt] = __builtin_amdgcn_wmma_f32_16x16x4_f32(
              false, afrag, false, bfrag, (short)0, acc[mt], false, false);
        }
      }
      if (w == 0 && ks + 1 < NKS) __builtin_amdgcn_s_wait_tensorcnt(0);
      __syncthreads();
    }

    // dist = sq_i - 2*s_ij + sq_j  -> stage 64x128 tile in LDS (aliases Bs)
    float sqj = sq[gb + subcol + w * 16 + ln];
#pragma unroll
    for (int mt = 0; mt < 4; ++mt) {
#pragma unroll
      for (int r = 0; r < 8; ++r) {
        int i_loc = mt * 16 + r + 8 * hi;  // C layout: VGPR r -> M=r (+8 hi half)
        float d = sqs[i_loc] - 2.0f * acc[mt][r] + sqj;
        Ds[i_loc * LDD + w * 16 + ln] = d;
      }
    }
    __syncthreads();

    // 4 threads per row scan 32 columns each, private sorted top-10 registers
    {
      int row = t >> 2, q = t & 3;
#pragma unroll 1
      for (int e = 0; e < 32; ++e) {
        int col = q * 32 + e;
        float d = Ds[row * LDD + col];
        int   id = subcol + col;
#pragma unroll
        for (int s = 0; s < KK; ++s) {
          if (d < bd[s]) {
            float td = bd[s]; bd[s] = d; d = td;
            int   ti = bi[s]; bi[s] = id; id = ti;
          }
        }
      }
    }
    __syncthreads();
  }

  // Merge the 4 per-row lists -> 10 candidates per row for this chunk
  {
    int row = t >> 2, q = t & 3;
#pragma unroll
    for (int s = 0; s < KK; ++s) {
      Md[(row * 4 + q) * KK + s] = bd[s];
      Mi[(row * 4 + q) * KK + s] = bi[s];
    }
    __syncthreads();
    if (q == 0) {
#pragma unroll 1
      for (int o = 1; o < 4; ++o) {
        for (int s2 = 0; s2 < KK; ++s2) {
          float d  = Md[(row * 4 + o) * KK + s2];
          int   id = Mi[(row * 4 + o) * KK + s2];
#pragma unroll
          for (int s = 0; s < KK; ++s) {
            bool better = (d < bd[s]) || (d == bd[s] && id < bi[s]);
            if (better) {
              float td = bd[s]; bd[s] = d; d = td;
              int   ti = bi[s]; bi[s] = id; id = ti;
            }
          }
        }
      }
      long base = ((gb + rowbase + row) * NCHUNK + chunk) * KK;
#pragma unroll
      for (int s = 0; s < KK; ++s) {
        cand_d[base + s] = bd[s];
        cand_i[base + s] = bi[s];
      }
    }
  }
}

// -------- Kernel 3: merge 8 chunk candidate lists, drop self, emit edges -----
__global__ void __launch_bounds__(256) knn_finalize(const float* __restrict__ cand_d,
                                                    const int*   __restrict__ cand_i,
                                                    long long*   __restrict__ out) {
  int g = blockIdx.x * blockDim.x + threadIdx.x;  // global row 0..B*N-1
  if (g >= B_ * N_) return;
  int b = g / N_;
  int n = g - b * N_;
  float bd[KK];
  int   bi[KK];
#pragma unroll
  for (int s = 0; s < KK; ++s) { bd[s] = 3.4e38f; bi[s] = 0x7fffffff; }
#pragma unroll 1
  for (int c = 0; c < NCHUNK * KK; ++c) {
    float d  = cand_d[(long)g * NCHUNK * KK + c];
    int   id = cand_i[(long)g * NCHUNK * KK + c];
#pragma unroll
    for (int s = 0; s < KK; ++s) {
      bool better = (d < bd[s]) || (d == bd[s] && id < bi[s]);
      if (better) {
        float td = bd[s]; bd[s] = d; d = td;
        int   ti = bi[s]; bi[s] = id; id = ti;
      }
    }
  }
  const long BNK = (long)B_ * N_ * K_;
#pragma unroll
  for (int k = 1; k < KK; ++k) {            // drop entry 0 (self)
    long e = (long)g * K_ + (k - 1);
    out[e]       = (long long)(bi[k] + b * N_);   // nn_flat
    out[BNK + e] = (long long)(n + b * N_);       // center_flat
  }
}

// ------------------------------- launcher ------------------------------------
extern "C" void kernel_launch(void* const* d_in, const int* in_sizes, int n_in,
                              void* d_out, int out_size, void* d_ws, size_t ws_size,
                              hipStream_t stream) {
  const float* x = (const float*)d_in[0];
  long long* out = (long long*)d_out;

  // Workspace layout (bytes):
  //   xn     : B*N*C*4            = 16,777,216
  //   sq     : B*N*4              =     65,536
  //   cand_d : B*N*NCHUNK*KK*4    =  5,242,880
  //   cand_i : B*N*NCHUNK*KK*4    =  5,242,880
  char* ws = (char*)d_ws;
  float* xn     = (float*)(ws);
  float* sq     = (float*)(ws + 16777216);
  float* cand_d = (float*)(ws + 16777216 + 65536);
  int*   cand_i = (int*)  (ws + 16777216 + 65536 + 5242880);

  knn_normalize<<<B_ * N_, 256, 0, stream>>>(x, xn, sq);

  dim3 grid2(N_ / MT, NCHUNK, B_);  // 128 x 8 x 2
  knn_gemm_topk<<<grid2, 256, 0, stream>>>(xn, sq, cand_d, cand_i);

  knn_finalize<<<(B_ * N_ + 255) / 256, 256, 0, stream>>>(cand_d, cand_i, out);
}